// SketchRenderNet_36223754175159
// MI455X (gfx1250) — compile-verified
//
#include <hip/hip_runtime.h>

typedef __attribute__((ext_vector_type(2))) float v2f;
typedef __attribute__((ext_vector_type(8))) float v8f;

#define IMSIZE   224
#define NSEG     1024
#define SMOOTH_K 8.0f
#define EPS_F    1e-10f
#define TILES_PER_ROW (IMSIZE / 16)   // 14
#define WAVES_PER_BLOCK 8
#define NUM_TILES ((IMSIZE * IMSIZE) / 16)  // 3136

// exp(-k*d) == exp2(d * (-k*log2(e)))
#define NEGK_LOG2E (-SMOOTH_K * 1.44269504088896340736f)
// out = -log(res/N)/k == (log2(N) - log2(res)) * (ln2/k)
#define LOG2_NSEG  10.0f
#define LN2_OVER_K (0.69314718055994530942f / SMOOTH_K)

__global__ __launch_bounds__(256) void sketch_render_kernel(
    const float* __restrict__ xpar,   // [1025] normalized x
    const float* __restrict__ ypar,   // [1025] normalized y
    float* __restrict__ out)          // [224*224]
{
    // Per-segment derived constants, computed once per block.
    __shared__ float4 sAB[NSEG];   // {A, B, Cc, slope}  : cx = A*yp + B*xp + Cc
    __shared__ float4 sQ[NSEG];    // {icept, minx, maxx, miny}
    __shared__ float2 sR[NSEG];    // {maxy, vert_flag}

    for (int s = threadIdx.x; s < NSEG; s += blockDim.x) {
        float x0 = xpar[s]     * (float)IMSIZE;
        float y0 = ypar[s]     * (float)IMSIZE;
        float x1 = xpar[s + 1] * (float)IMSIZE;
        float y1 = ypar[s + 1] * (float)IMSIZE;
        float dx = x1 - x0, dy = y1 - y0;
        float d2 = dx * dx + dy * dy;
        float inv = 1.0f / (d2 + EPS_F);
        float cross = y0 * x1 - x0 * y1;
        float A  = dx * dy * inv;
        float B  = dx * dx * inv;
        float Cc = -dy * cross * inv;
        float idx = 1.0f / (dx + EPS_F);
        float sl = dy * idx;
        float ic = cross * idx;
        float vert = (x0 == x1) ? 1.0f : 0.0f;
        sAB[s] = make_float4(A, B, Cc, sl);
        sQ[s]  = make_float4(ic, fminf(x0, x1), fmaxf(x0, x1), fminf(y0, y1));
        sR[s]  = make_float2(fmaxf(y0, y1), vert);
    }
    __syncthreads();

    // One wave = one 16-pixel tile (16 consecutive pixels of one image row).
    int wid  = (blockIdx.x * blockDim.x + threadIdx.x) >> 5;  // tile id
    int lane = threadIdx.x & 31;
    int grp  = lane >> 4;    // lane group: K-slots {0,1} or {2,3}
    int m    = lane & 15;    // A-matrix row M = pixel within tile
    int row  = wid / TILES_PER_ROW;
    int col  = wid % TILES_PER_ROW;
    float xp = (float)(col * 16 + m);
    float yp = (float)row;

    v8f c = {0.0f, 0.0f, 0.0f, 0.0f, 0.0f, 0.0f, 0.0f, 0.0f};
    v2f bones = {1.0f, 1.0f};                 // B = all-ones 4x16

    #pragma unroll 2
    for (int s = 0; s < NSEG; s += 4) {
        int k0 = s + grp * 2;
        v2f a;
        #pragma unroll
        for (int j = 0; j < 2; ++j) {
            int k = k0 + j;
            float4 p = sAB[k];
            float4 q = sQ[k];
            float2 r = sR[k];
            // closest point on infinite line
            float cx = fmaf(p.x, yp, fmaf(p.y, xp, p.z));
            float cy = fmaf(p.w, cx, q.x);
            // vertical special case: yp1 <- yp (xp1 handled by bbox clamp)
            float yps = (r.y != 0.0f) ? yp : cy;
            // clamp to segment bounding box
            float xp1 = fminf(fmaxf(cx,  q.y), q.z);
            float yp1 = fminf(fmaxf(yps, q.w), r.x);
            float ddx = xp - xp1;
            float ddy = yp - yp1;
            float t = fmaf(ddx, ddx, EPS_F);
            t = fmaf(ddy, ddy, t);
            float dist = __fsqrt_rn(t);             // FUZZ == 1
            // single mul feeding v_exp_f32
            a[j] = __builtin_amdgcn_exp2f(dist * NEGK_LOG2E);
        }
        // D[m][n] += sum_k A[m][k] * 1 : matrix pipe does the reduction in f32.
        c = __builtin_amdgcn_wmma_f32_16x16x4_f32(false, a, false, bones,
                                                  (short)0, c, false, false);
    }

    // All N columns of D are identical (B was ones).
    // Lanes 0-15 hold M=0..7 in c[0..7]; lanes 16-31 hold M=8..15.
    if (lane == 0 || lane == 16) {
        int base = row * IMSIZE + col * 16 + grp * 8;
        #pragma unroll
        for (int rI = 0; rI < 8; ++rI) {
            float res = c[rI];
            // raw v_log_f32; res in (0, 1024], no denorm concerns
            float val = (LOG2_NSEG - __builtin_amdgcn_logf(res)) * LN2_OVER_K;
            out[base + rI] = val;
        }
    }
}

extern "C" void kernel_launch(void* const* d_in, const int* in_sizes, int n_in,
                              void* d_out, int out_size, void* d_ws, size_t ws_size,
                              hipStream_t stream) {
    (void)in_sizes; (void)n_in; (void)out_size; (void)d_ws; (void)ws_size;
    const float* x = (const float*)d_in[0];
    const float* y = (const float*)d_in[1];
    float* out = (float*)d_out;
    dim3 grid(NUM_TILES / WAVES_PER_BLOCK);   // 392 blocks
    dim3 block(WAVES_PER_BLOCK * 32);         // 256 threads = 8 wave32
    sketch_render_kernel<<<grid, block, 0, stream>>>(x, y, out);
}